// ModuleRenderRT_53850299957899
// MI455X (gfx1250) — compile-verified
//
#include <hip/hip_runtime.h>
#include <stdint.h>

#define WIDTH  1024
#define HEIGHT 1024
#define HW     (WIDTH*HEIGHT)
#define CCH    4
#define TILE   16
#define HALO   22                 // max |r| < 21 px (K<30, |coff-df|<0.7) + rounding
#define HMAX   (TILE + 2*HALO)    // 60

// ---------------- compile-time aperture table (samples_per_side = 17) ----------------
#define SPS 17
constexpr int count_samples() {
  int n = 0;
  for (int i = 0; i < SPS; ++i)
    for (int j = 0; j < SPS; ++j) {
      float x = -1.0f + 0.125f * i, y = -1.0f + 0.125f * j;
      if (x * x + y * y <= 1.0f) ++n;
    }
  return n;
}
constexpr int NS = count_samples();   // 197

constexpr double csqrt_(double v) {
  if (v <= 0.0) return 0.0;
  double g = v < 1.0 ? 1.0 : v;
  for (int k = 0; k < 100; ++k) g = 0.5 * (g + v / g);
  return g;
}

struct SampleTbl { float v[NS][4]; };
constexpr SampleTbl make_tbl() {
  SampleTbl t{};
  int n = 0;
  for (int i = 0; i < SPS; ++i)
    for (int j = 0; j < SPS; ++j) {
      float x = -1.0f + 0.125f * i, y = -1.0f + 0.125f * j;
      if (x * x + y * y <= 1.0f) {
        t.v[n][0] = x;
        t.v[n][1] = y;
        t.v[n][2] = (float)csqrt_((double)x * (double)x + (double)y * (double)y);
        t.v[n][3] = 0.0f;
        ++n;
      }
    }
  return t;
}
__device__ __constant__ SampleTbl g_tbl = make_tbl();

// ---------------- TDM: 2D tile DMA global -> LDS (CDNA5 Tensor Data Mover) ----------------
typedef uint32_t v4u __attribute__((ext_vector_type(4)));
typedef int      v8i __attribute__((ext_vector_type(8)));
typedef int      v4i __attribute__((ext_vector_type(4)));

// tile_w8 / stride8 are in 8-byte elements (data_size = 3). Tensor dims set exactly
// equal to the (pre-clamped) tile so no OOB handling is triggered.
__device__ __forceinline__ void tdm_load_tile_2d(uint32_t lds_addr, const void* global_tile,
                                                 uint32_t tile_w8, uint32_t tile_h,
                                                 uint32_t stride8) {
  uint64_t ga = (uint64_t)(uintptr_t)global_tile;
  v4u g0;
  g0[0] = 1u;                                                  // count=1 (valid), user mode
  g0[1] = lds_addr;                                            // LDS byte address
  g0[2] = (uint32_t)ga;                                        // global_addr[31:0]
  g0[3] = ((uint32_t)(ga >> 32) & 0x01FFFFFFu) | (2u << 30);   // global_addr[56:32] | type=2
  v8i g1;
  g1[0] = (int)(3u << 16);                         // data_size=3 (8B); no multicast/pad/iterate
  g1[1] = (int)(tile_w8 << 16);                    // tensor_dim0[15:0]
  g1[2] = (int)((tile_w8 >> 16) | (tile_h << 16)); // tensor_dim0[31:16] | tensor_dim1[15:0]
  g1[3] = (int)((tile_h >> 16) | (tile_w8 << 16)); // tensor_dim1[31:16] | tile_dim0
  g1[4] = (int)(tile_h & 0xFFFFu);                 // tile_dim1 | tile_dim2=0
  g1[5] = (int)stride8;                            // tensor_dim0_stride[31:0]
  g1[6] = 0;                                       // stride0[47:32] | stride1[15:0]
  g1[7] = 0;                                       // stride1[47:16]
  v4i gz = {0, 0, 0, 0};
#if __clang_major__ >= 23
  v8i gz8 = {0, 0, 0, 0, 0, 0, 0, 0};
  __builtin_amdgcn_tensor_load_to_lds(g0, g1, gz, gz, gz8, 0);
#else
  __builtin_amdgcn_tensor_load_to_lds(g0, g1, gz, gz, 0);
#endif
}

// ---------------- pass 1: pack per-pixel (r, g0) and g0*img (streams HBM/L2 once) ----------------
__global__ __launch_bounds__(256) void prep_kernel(
    const float* __restrict__ images, const float* __restrict__ alphas,
    const float* __restrict__ coffs, const float* __restrict__ Kv,
    const float* __restrict__ dfv,
    float2* __restrict__ rg, float4* __restrict__ g4) {
  const int p = blockIdx.x * blockDim.x + threadIdx.x;
  const int b = blockIdx.y;
  const size_t ip = (size_t)b * HW + p;
  const float K = Kv[b], df = dfv[b];
  const float r = K * (coffs[ip] - df);
  const float a = alphas[ip];
  const float g0 = a / (3.14159265358979323846f * r * r + 1.0f);
  rg[ip] = make_float2(r, g0);
  const float* im = images + (size_t)b * CCH * HW + p;
  g4[ip] = make_float4(g0 * im[0], g0 * im[(size_t)HW],
                       g0 * im[2 * (size_t)HW], g0 * im[3 * (size_t)HW]);
}

// ---------------- pass 2: TDM-staged LDS gather + accumulate ----------------
__global__ __launch_bounds__(256) void render_kernel(
    const float2* __restrict__ rg, const float4* __restrict__ g4,
    const float* __restrict__ Eta, float* __restrict__ out, int B) {
  extern __shared__ char smem[];
  const int tx0 = blockIdx.x * TILE, ty0 = blockIdx.y * TILE, b = blockIdx.z;
  int hx0 = tx0 - HALO; if (hx0 < 0) hx0 = 0;
  int hy0 = ty0 - HALO; if (hy0 < 0) hy0 = 0;
  int hx1 = tx0 + TILE - 1 + HALO; if (hx1 > WIDTH - 1)  hx1 = WIDTH - 1;
  int hy1 = ty0 + TILE - 1 + HALO; if (hy1 > HEIGHT - 1) hy1 = HEIGHT - 1;
  const int hw = hx1 - hx0 + 1, hh = hy1 - hy0 + 1;
  const uint32_t rgBytes = (uint32_t)(hw * hh) * 8u;
  const uint32_t lds0 = (uint32_t)(uintptr_t)&smem[0];   // low 32 bits = LDS byte offset

  if (threadIdx.y < 2) {  // first wave32 issues the TDM DMAs, waits on TENSORcnt
    const float2* srcRG = rg + ((size_t)b * HW + (size_t)hy0 * WIDTH + hx0);
    const float4* srcG4 = g4 + ((size_t)b * HW + (size_t)hy0 * WIDTH + hx0);
    tdm_load_tile_2d(lds0,           srcRG, (uint32_t)hw,       (uint32_t)hh, WIDTH);
    tdm_load_tile_2d(lds0 + rgBytes, srcG4, (uint32_t)(2 * hw), (uint32_t)hh, 2 * WIDTH);
    __builtin_amdgcn_s_wait_tensorcnt(0);
  }
  __syncthreads();

  const float2* __restrict__ Lrg = (const float2*)smem;
  const float4* __restrict__ Lg4 = (const float4*)(smem + rgBytes);
  const int x = tx0 + threadIdx.x, y = ty0 + threadIdx.y;
  const float eta = Eta[b];
  const float2 self = Lrg[(y - hy0) * hw + (x - hx0)];
  const float r = self.x;
  const float er = eta * fabsf(r);          // eta * r_abs, hoisted
  const float fx = (float)x, fy = (float)y;
  float4 acc = make_float4(0.f, 0.f, 0.f, 0.f);
  float wc = 0.f;

#pragma unroll 4
  for (int s = 0; s < NS; ++s) {
    const float xs = g_tbl.v[s][0], ys = g_tbl.v[s][1], len = g_tbl.v[s][2];
    // round-half-even + clip, matching jnp.round / jnp.clip.  The clipped global
    // coordinate is provably inside the clamped halo box (|r| < HALO), so the
    // local coordinate needs no further clamping.
    const float sxf = fminf(fmaxf(rintf(fmaf(r, xs, fx)), 0.f), (float)(WIDTH - 1));
    const float syf = fminf(fmaxf(rintf(fmaf(r, ys, fy)), 0.f), (float)(HEIGHT - 1));
    const int gx = (int)sxf - hx0;
    const int gy = (int)syf - hy0;
    const int off = gy * hw + gx;
    const float2 q = Lrg[off];        // ds_load_b64: (r_q, g0_q)
    const float4 G = Lg4[off];        // ds_load_b128: g0_q * img_q
    const float t0 = fmaf(-er, len, 1.0f);            // 1 - eta*r_abs*len
    const float t  = fmaf(eta, fabsf(q.x), t0);       // eta*(r_q - dist) + 1
    const float sig = __builtin_amdgcn_rcpf(1.0f + __expf(-t));   // sigmoid
    wc    = fmaf(sig, q.y, wc);
    acc.x = fmaf(sig, G.x, acc.x);
    acc.y = fmaf(sig, G.y, acc.y);
    acc.z = fmaf(sig, G.z, acc.z);
    acc.w = fmaf(sig, G.w, acc.w);
  }

  const size_t p = (size_t)y * WIDTH + x;
  float* ob = out + ((size_t)b * CCH) * HW + p;
  ob[0] = acc.x;
  ob[(size_t)HW] = acc.y;
  ob[2 * (size_t)HW] = acc.z;
  ob[3 * (size_t)HW] = acc.w;
  out[(size_t)B * CCH * HW + (size_t)b * HW + p] = wc;   // wc after concatenated bok
}

// ---------------- fallback: direct L2 gathers (used only if d_ws is too small) ----------------
__global__ __launch_bounds__(256) void render_direct(
    const float* __restrict__ images, const float* __restrict__ alphas,
    const float* __restrict__ coffs, const float* __restrict__ Kv,
    const float* __restrict__ dfv, const float* __restrict__ Eta,
    float* __restrict__ out, int B) {
  const int p = blockIdx.x * blockDim.x + threadIdx.x;
  const int b = blockIdx.y;
  const int x = p & (WIDTH - 1), y = p >> 10;
  const float K = Kv[b], df = dfv[b], eta = Eta[b];
  const size_t ip = (size_t)b * HW + p;
  const float r = K * (coffs[ip] - df);
  const float er = eta * fabsf(r);
  const float fx = (float)x, fy = (float)y;
  const float* cf = coffs + (size_t)b * HW;
  const float* al = alphas + (size_t)b * HW;
  const float* im = images + (size_t)b * CCH * HW;
  float4 acc = make_float4(0.f, 0.f, 0.f, 0.f);
  float wc = 0.f;
#pragma unroll 2
  for (int s = 0; s < NS; ++s) {
    const float xs = g_tbl.v[s][0], ys = g_tbl.v[s][1], len = g_tbl.v[s][2];
    const float sxf = fminf(fmaxf(rintf(fmaf(r, xs, fx)), 0.f), (float)(WIDTH - 1));
    const float syf = fminf(fmaxf(rintf(fmaf(r, ys, fy)), 0.f), (float)(HEIGHT - 1));
    const int idx = ((int)syf << 10) + (int)sxf;
    const float rq = fabsf(K * (cf[idx] - df));
    const float g0q = al[idx] * __builtin_amdgcn_rcpf(fmaf(3.14159265358979323846f * rq, rq, 1.0f));
    const float t = fmaf(eta, rq, fmaf(-er, len, 1.0f));
    const float sig = __builtin_amdgcn_rcpf(1.0f + __expf(-t));
    const float w = sig * g0q;
    wc    += w;
    acc.x = fmaf(w, im[idx], acc.x);
    acc.y = fmaf(w, im[idx + (size_t)HW], acc.y);
    acc.z = fmaf(w, im[idx + 2 * (size_t)HW], acc.z);
    acc.w = fmaf(w, im[idx + 3 * (size_t)HW], acc.w);
  }
  float* ob = out + ((size_t)b * CCH) * HW + p;
  ob[0] = acc.x;
  ob[(size_t)HW] = acc.y;
  ob[2 * (size_t)HW] = acc.z;
  ob[3 * (size_t)HW] = acc.w;
  out[(size_t)B * CCH * HW + ip] = wc;
}

// ---------------- host launch ----------------
extern "C" void kernel_launch(void* const* d_in, const int* in_sizes, int n_in,
                              void* d_out, int out_size, void* d_ws, size_t ws_size,
                              hipStream_t stream) {
  const float* images = (const float*)d_in[0];
  const float* alphas = (const float*)d_in[1];
  const float* coffs  = (const float*)d_in[2];
  const float* Kv     = (const float*)d_in[3];
  const float* dfv    = (const float*)d_in[4];
  const float* eta    = (const float*)d_in[5];
  const int B = in_sizes[3];   // K has shape [B,1,1,1]

  const size_t need = (size_t)B * HW * (sizeof(float2) + sizeof(float4));
  if (ws_size >= need) {
    float2* rg = (float2*)d_ws;
    float4* g4 = (float4*)((char*)d_ws + (size_t)B * HW * sizeof(float2));

    dim3 pb(256), pg(HW / 256, B);
    prep_kernel<<<pg, pb, 0, stream>>>(images, alphas, coffs, Kv, dfv, rg, g4);

    const uint32_t shmem = (uint32_t)(HMAX * HMAX * (sizeof(float2) + sizeof(float4))); // 86400 B
    hipFuncSetAttribute(reinterpret_cast<const void*>(render_kernel),
                        hipFuncAttributeMaxDynamicSharedMemorySize, (int)shmem);
    dim3 rb(TILE, TILE), rgrid(WIDTH / TILE, HEIGHT / TILE, B);
    render_kernel<<<rgrid, rb, shmem, stream>>>(rg, g4, eta, (float*)d_out, B);
  } else {
    dim3 db(256), dg(HW / 256, B);
    render_direct<<<dg, db, 0, stream>>>(images, alphas, coffs, Kv, dfv, eta,
                                         (float*)d_out, B);
  }
}